// GNN_52896817217884
// MI455X (gfx1250) — compile-verified
//
#include <hip/hip_runtime.h>

typedef __attribute__((ext_vector_type(2))) float v2f;
typedef __attribute__((ext_vector_type(8))) float v8f;

// ---------------------------------------------------------------------------
// K1: deg[i] = 1.0 (self loop contribution)
// ---------------------------------------------------------------------------
__global__ void k_init_deg(float* __restrict__ deg, int n) {
  int i = blockIdx.x * blockDim.x + threadIdx.x;
  if (i < n) deg[i] = 1.0f;
}

// ---------------------------------------------------------------------------
// K2: deg[dst] += 1 for every edge
// ---------------------------------------------------------------------------
__global__ void k_deg_edges(const long long* __restrict__ ei,
                            float* __restrict__ deg, int E) {
  int e = blockIdx.x * blockDim.x + threadIdx.x;
  if (e < E) {
    int d = (int)ei[(long long)E + e];
    atomicAdd(&deg[d], 1.0f);
  }
}

// ---------------------------------------------------------------------------
// K3: dinv = rsqrt(deg) (in place), s[i] = dinv^2 * x[i] (self-loop msg)
// ---------------------------------------------------------------------------
__global__ void k_dinv_sinit(float* __restrict__ deg_dinv,
                             const float* __restrict__ x,
                             float* __restrict__ s, int n) {
  int i = blockIdx.x * blockDim.x + threadIdx.x;
  if (i < n) {
    float di = rsqrtf(deg_dinv[i]);   // deg >= 1 always (self loops)
    deg_dinv[i] = di;
    s[i] = di * di * x[i];
  }
}

// ---------------------------------------------------------------------------
// K4: layer-1 scalar scatter: s[dst] += dinv[src]*dinv[dst]*x[src]
// ---------------------------------------------------------------------------
__global__ void k_layer1_edges(const long long* __restrict__ ei,
                               const float* __restrict__ dinv,
                               const float* __restrict__ x,
                               float* __restrict__ s, int E) {
  int e = blockIdx.x * blockDim.x + threadIdx.x;
  if (e < E) {
    int si = (int)ei[e];
    int di = (int)ei[(long long)E + e];
    float norm = dinv[si] * dinv[di];
    atomicAdd(&s[di], norm * x[si]);
  }
}

// ---------------------------------------------------------------------------
// K5: per-node transform with fp32 WMMA.
//   h[i,j]  = relu(W1[j]*s[i] + b1[j])          (16 features, rank-1 layer 1)
//   g[i,k]  = sum_j h[i,j]*W2[j,k]              (16x16 h-tile @ 16x16 pad(W2))
//   out[i,k]= dinv[i]^2 * g[i,k] + b2[k]        (self-loop init for layer 2)
// One wave handles 16 nodes; K=16 split into 4 chained V_WMMA_F32_16X16X4_F32.
// A layout (16x4 f32): lane m holds K={4t,4t+1}; lane m+16 holds K={4t+2,4t+3}.
// B layout (4x16 f32): v.x = row(4t+2*half) col(lane&15); v.y = next row.
// D layout (16x16 f32): element [M = r + 8*half, N = lane&15] in c[r].
// ---------------------------------------------------------------------------
__global__ void k_node_transform(const float* __restrict__ s,
                                 const float* __restrict__ dinv,
                                 const float* __restrict__ W1,
                                 const float* __restrict__ b1,
                                 const float* __restrict__ W2,
                                 const float* __restrict__ b2,
                                 float* __restrict__ g,
                                 float* __restrict__ out, int n) {
  int lane = threadIdx.x & 31;
  int wave = (blockIdx.x * blockDim.x + threadIdx.x) >> 5;
  int nodeBase = wave * 16;
  int m    = lane & 15;
  int half = lane >> 4;

  int node  = nodeBase + m;
  int nodeC = node < n ? node : (n - 1);   // clamp loads; keep EXEC all-1s
  float sv = s[nodeC];

  int col = m;                              // B/D column for this lane
  v8f c = {};
#pragma unroll
  for (int t = 0; t < 4; ++t) {
    int j0 = 4 * t + 2 * half;             // feature / K index pair
    v2f a, b;
    a.x = fmaxf(fmaf(W1[j0],     sv, b1[j0]),     0.0f);
    a.y = fmaxf(fmaf(W1[j0 + 1], sv, b1[j0 + 1]), 0.0f);
    b.x = (col < 2) ? W2[j0 * 2 + col]       : 0.0f;
    b.y = (col < 2) ? W2[(j0 + 1) * 2 + col] : 0.0f;
    c = __builtin_amdgcn_wmma_f32_16x16x4_f32(
        /*neg_a=*/false, a, /*neg_b=*/false, b,
        /*c_mod=*/(short)0, c, /*reuse_a=*/false, /*reuse_b=*/false);
  }

  if (col < 2) {
#pragma unroll
    for (int r = 0; r < 8; ++r) {
      int nd = nodeBase + r + 8 * half;    // M = r + 8*half
      if (nd < n) {
        float gv = c[r];
        g[nd * 2 + col] = gv;
        float di = dinv[nd];
        out[nd * 2 + col] = fmaf(di * di, gv, b2[col]);
      }
    }
  }
}

// ---------------------------------------------------------------------------
// K6: layer-2 scatter: out[dst,k] += dinv[src]*dinv[dst]*g[src,k], k=0,1
// ---------------------------------------------------------------------------
__global__ void k_layer2_edges(const long long* __restrict__ ei,
                               const float* __restrict__ dinv,
                               const float* __restrict__ g,
                               float* __restrict__ out, int E) {
  int e = blockIdx.x * blockDim.x + threadIdx.x;
  if (e < E) {
    int si = (int)ei[e];
    int di = (int)ei[(long long)E + e];
    float norm = dinv[si] * dinv[di];
    atomicAdd(&out[di * 2 + 0], norm * g[si * 2 + 0]);
    atomicAdd(&out[di * 2 + 1], norm * g[si * 2 + 1]);
  }
}

// ---------------------------------------------------------------------------
extern "C" void kernel_launch(void* const* d_in, const int* in_sizes, int n_in,
                              void* d_out, int out_size, void* d_ws, size_t ws_size,
                              hipStream_t stream) {
  const float*     x  = (const float*)d_in[0];
  const long long* ei = (const long long*)d_in[1];   // int64 [2, E] flat
  const float*     W1 = (const float*)d_in[2];       // [1,16]
  const float*     b1 = (const float*)d_in[3];       // [16]
  const float*     W2 = (const float*)d_in[4];       // [16,2]
  const float*     b2 = (const float*)d_in[5];       // [2]
  float* out = (float*)d_out;                        // [N,2]

  const int N = in_sizes[0];       // x is [N,1]
  const int E = in_sizes[1] / 2;   // edge_index flat = 2*E

  // workspace: dinv (N) | s (N) | g (2N)   -> 4N floats
  float* dinv = (float*)d_ws;
  float* s    = dinv + N;
  float* g    = s + N;

  const int T = 256;
  const int gN = (N + T - 1) / T;
  const int gE = (E + T - 1) / T;
  const int nodesPerBlock = (T / 32) * 16;            // 8 waves * 16 nodes
  const int gT = (N + nodesPerBlock - 1) / nodesPerBlock;

  k_init_deg      <<<gN, T, 0, stream>>>(dinv, N);
  k_deg_edges     <<<gE, T, 0, stream>>>(ei, dinv, E);
  k_dinv_sinit    <<<gN, T, 0, stream>>>(dinv, x, s, N);
  k_layer1_edges  <<<gE, T, 0, stream>>>(ei, dinv, x, s, E);
  k_node_transform<<<gT, T, 0, stream>>>(s, dinv, W1, b1, W2, b2, g, out, N);
  k_layer2_edges  <<<gE, T, 0, stream>>>(ei, dinv, g, out, E);
}